// GNNClassifier_8022998909728
// MI455X (gfx1250) — compile-verified
//
#include <hip/hip_runtime.h>

typedef __attribute__((ext_vector_type(2))) float v2f;
typedef __attribute__((ext_vector_type(8))) float v8f;

#define K_F 64  // input feature dim of both layers (IN_F == HID == 64)

// ---------------------------------------------------------------------------
// Utility: zero a float buffer (scratch must be re-zeroed every launch since
// the aggregation uses atomics).
// ---------------------------------------------------------------------------
__global__ void zero_kernel(float* __restrict__ p, int n) {
  int i = blockIdx.x * blockDim.x + threadIdx.x;
  if (i < n) p[i] = 0.0f;
}

// ---------------------------------------------------------------------------
// Degree: one thread per edge, count in-degree of dst.
// ---------------------------------------------------------------------------
__global__ void degree_kernel(const int* __restrict__ dst,
                              float* __restrict__ deg, int nE) {
  int i = blockIdx.x * blockDim.x + threadIdx.x;
  if (i < nE) atomicAdd(&deg[dst[i]], 1.0f);
}

// deg[i] -> 1 / max(deg[i], 1)
__global__ void inv_kernel(float* __restrict__ deg, int n) {
  int i = blockIdx.x * blockDim.x + threadIdx.x;
  if (i < n) deg[i] = 1.0f / fmaxf(deg[i], 1.0f);
}

// ---------------------------------------------------------------------------
// Scatter: one wave32 per edge; lane l carries features (2l, 2l+1) as a
// float2 load from feat[src] and two global f32 atomic adds into msg[dst].
// ---------------------------------------------------------------------------
__global__ void scatter_kernel(const float* __restrict__ feat,
                               const int* __restrict__ src,
                               const int* __restrict__ dst,
                               float* __restrict__ msg, int nE) {
  int edge = blockIdx.x * (blockDim.x >> 5) + (threadIdx.x >> 5);
  if (edge >= nE) return;                    // wave-uniform exit
  int lane = threadIdx.x & 31;
  int s = src[edge];
  int d = dst[edge];
  v2f v = *reinterpret_cast<const v2f*>(feat + (size_t)s * K_F + lane * 2);
  float* base = msg + (size_t)d * K_F + lane * 2;
  atomicAdd(base, v.x);
  atomicAdd(base + 1, v.y);
}

// ---------------------------------------------------------------------------
// Fused SAGE layer GEMM: out = act( (msg .* invdeg) @ Wl + xin @ Wr + bias )
// One wave32 computes one 16x16 output tile with V_WMMA_F32_16X16X4_F32.
//
// A layout (16x4 f32):  lane<16 holds row M=lane, K=(k,k+1);
//                       lane>=16 holds row M=lane-16, K=(k+2,k+3).
// B layout (4x16 f32):  VGPR0/1 hold K rows (koff, koff+1) at N=lane&15.
// C layout (16x16 f32): VGPR i holds M = i + 8*(lane>=16), N = lane&15.
// ---------------------------------------------------------------------------
template <int OUT, bool RELU>
__global__ void sage_gemm(const float* __restrict__ msg,
                          const float* __restrict__ invdeg,
                          const float* __restrict__ xin,
                          const float* __restrict__ Wl,
                          const float* __restrict__ Wr,
                          const float* __restrict__ bias,
                          float* __restrict__ out,
                          int nTilesN, int nTiles) {
  int tile = blockIdx.x * (blockDim.x >> 5) + (threadIdx.x >> 5);
  if (tile >= nTiles) return;                // wave-uniform exit, EXEC stays ~0
  int tm = tile / nTilesN;
  int tn = tile - tm * nTilesN;

  int lane = threadIdx.x & 31;
  int hrow = lane & 15;                      // 0..15
  int hi   = lane >> 4;                      // half-wave select
  int row  = tm * 16 + hrow;                 // node index (A row, and C row base)
  int col  = tn * 16 + hrow;                 // output feature (B/C column)
  int koff = hi * 2;                         // K sub-offset within a K=4 step

  float inv = invdeg[row];
  const float* mrow = msg + (size_t)row * K_F + koff;
  const float* xrow = xin + (size_t)row * K_F + koff;

  v8f c = {};
#pragma unroll
  for (int k = 0; k < K_F; k += 4) {
    v2f am = *reinterpret_cast<const v2f*>(mrow + k);   // mean-aggregate path
    am.x *= inv;
    am.y *= inv;
    v2f ax = *reinterpret_cast<const v2f*>(xrow + k);   // self path
    v2f bl, br;
    bl.x = Wl[(k + koff) * OUT + col];
    bl.y = Wl[(k + koff + 1) * OUT + col];
    br.x = Wr[(k + koff) * OUT + col];
    br.y = Wr[(k + koff + 1) * OUT + col];
    // D = A*B + C, full fp32 (matches reference precision)
    c = __builtin_amdgcn_wmma_f32_16x16x4_f32(false, am, false, bl,
                                              (short)0, c, false, false);
    c = __builtin_amdgcn_wmma_f32_16x16x4_f32(false, ax, false, br,
                                              (short)0, c, false, false);
  }

  float bv = bias[col];
#pragma unroll
  for (int i = 0; i < 8; ++i) {
    float v = c[i] + bv;
    if (RELU) v = fmaxf(v, 0.0f);
    out[(size_t)(tm * 16 + hi * 8 + i) * OUT + col] = v;
  }
}

// ---------------------------------------------------------------------------
// Driver
// ---------------------------------------------------------------------------
extern "C" void kernel_launch(void* const* d_in, const int* in_sizes, int n_in,
                              void* d_out, int out_size, void* d_ws,
                              size_t ws_size, hipStream_t stream) {
  (void)n_in; (void)out_size; (void)ws_size;

  const float* x   = (const float*)d_in[0];
  const int*   ei  = (const int*)d_in[1];   // int32 per harness convention
  const float* W1l = (const float*)d_in[2];
  const float* b1  = (const float*)d_in[3];
  const float* W1r = (const float*)d_in[4];
  const float* W2l = (const float*)d_in[5];
  const float* b2  = (const float*)d_in[6];
  const float* W2r = (const float*)d_in[7];
  float* out = (float*)d_out;

  const int N  = in_sizes[0] / K_F;   // 50000 (divisible by 16)
  const int nE = in_sizes[1] / 2;     // 800000
  const int* src = ei;                // edge_index[0]
  const int* dst = ei + nE;           // edge_index[1]

  // Workspace layout: [msg: N*64][deg: N][h: N*64]
  float* msg = (float*)d_ws;
  float* deg = msg + (size_t)N * K_F;
  float* h   = deg + N;

  const int T = 256;                  // 8 waves per block

  // ---- shared: degree / inverse degree ----
  {
    int n0 = N * K_F + N;             // zero msg and deg together (contiguous)
    zero_kernel<<<(n0 + T - 1) / T, T, 0, stream>>>(msg, n0);
  }
  degree_kernel<<<(nE + T - 1) / T, T, 0, stream>>>(dst, deg, nE);
  inv_kernel<<<(N + T - 1) / T, T, 0, stream>>>(deg, N);

  // ---- layer 1: h = relu(mean(x) @ W1l + x @ W1r + b1) ----
  scatter_kernel<<<(nE + 7) / 8, T, 0, stream>>>(x, src, dst, msg, nE);
  {
    int nTilesN = K_F / 16;           // HID = 64 -> 4 column tiles
    int nTiles  = (N / 16) * nTilesN; // 12500
    int blocks  = (nTiles + 7) / 8;
    sage_gemm<K_F, true><<<blocks, T, 0, stream>>>(msg, deg, x, W1l, W1r, b1,
                                                   h, nTilesN, nTiles);
  }

  // ---- layer 2: out = mean(h) @ W2l + h @ W2r + b2 ----
  {
    int n1 = N * K_F;
    zero_kernel<<<(n1 + T - 1) / T, T, 0, stream>>>(msg, n1);
  }
  scatter_kernel<<<(nE + 7) / 8, T, 0, stream>>>(h, src, dst, msg, nE);
  {
    int nTilesN = 16 / 16;            // N_CLS = 16 -> 1 column tile
    int nTiles  = (N / 16) * nTilesN; // 3125
    int blocks  = (nTiles + 7) / 8;
    sage_gemm<16, false><<<blocks, T, 0, stream>>>(msg, deg, h, W2l, W2r, b2,
                                                   out, nTilesN, nTiles);
  }
}